// HungarianMatcher_base_48146583388352
// MI455X (gfx1250) — compile-verified
//
#include <hip/hip_runtime.h>
#include <hip/hip_bf16.h>
#include <float.h>

// Problem constants (match reference setup_inputs)
#define BB     32
#define QQ     900
#define QB     900
#define NCLS   91
#define NT     30
#define NROW   (BB * QB)        // 28800 pseudo-target rows
#define NLOGIT ((size_t)BB * QQ * NCLS)   // 2,620,800

#define ALPHA  0.25f
#define EPSF   1e-8f
#define W_CLS  2.0f
#define W_BOX  5.0f
#define W_GIOU 2.0f
#define BIGC   1e9f

#define JTILE  256
#define QCHUNK 60               // 15 chunks * 60 = 900 exactly

// ---------------- CDNA5-specific helpers (async global->LDS + waits) -------

typedef __attribute__((ext_vector_type(4))) int v4i_t;
typedef __attribute__((address_space(1))) v4i_t* gv4i_p;
typedef __attribute__((address_space(3))) v4i_t* lv4i_p;

__device__ __forceinline__ void async_copy_b128_to_lds(const void* gptr, void* lptr) {
#if __has_builtin(__builtin_amdgcn_global_load_async_to_lds_b128)
    __builtin_amdgcn_global_load_async_to_lds_b128(
        (gv4i_p)gptr,
        (lv4i_p)lptr,
        0, 0);
#else
    unsigned lds_off = (unsigned)(size_t)(__attribute__((address_space(3))) char*)lptr;
    asm volatile("global_load_async_to_lds_b128 %0, %1, off"
                 :: "v"(lds_off), "v"(gptr) : "memory");
#endif
}

__device__ __forceinline__ void wait_async_zero() {
#if __has_builtin(__builtin_amdgcn_s_wait_asynccnt)
    __builtin_amdgcn_s_wait_asynccnt(0);
#else
    asm volatile("s_wait_asynccnt 0" ::: "memory");
#endif
}

// Focal-style class cost (already scaled by W_CLS)
__device__ __forceinline__ float class_cost_from_logit(float x) {
    const float p   = 1.0f / (1.0f + expf(-x));
    const float omp = 1.0f - p;
    const float cls = ALPHA * omp * omp * (-logf(p + EPSF))
                    - (1.0f - ALPHA) * p * p * (-logf(omp + EPSF));
    return W_CLS * cls;
}

// ---------------- Kernel 1: pseudo-target prep ------------------------------
// One wave32 per (b, j) row: argmax/max over 91 logits (lane-strided +
// shfl_xor butterfly with first-index tiebreak), GIoU filter vs 30 targets
// (one target per lane, __all reduce), write SoA struct to workspace.
//
// ws layout (floats, stride NROW): [0]=cx [1]=cy [2]=w [3]=h
//   [4]=x0 [5]=y0 [6]=x1 [7]=y1 [8]=area [9]=label(int bits) [10]=mask(0/1)
__global__ __launch_bounds__(256) void prep_kernel(
    const float* __restrict__ logits_base,  // [B, QB, NCLS]
    const float* __restrict__ boxes_base,   // [B, QB, 4]
    const float* __restrict__ tboxes,       // [B, NT, 4]
    float* __restrict__ ws,
    float* __restrict__ out_mask)           // [B*QB]
{
    const int wave = threadIdx.x >> 5;
    const int lane = threadIdx.x & 31;
    const int row  = blockIdx.x * 8 + wave;
    if (row >= NROW) return;
    const int b = row / QB;

    // ---- max/argmax over classes (first max wins, like jnp.argmax) ----
    float m = -FLT_MAX; int mi = 0x7fffffff;
    const float* lrow = logits_base + (size_t)row * NCLS;
    for (int k = lane; k < NCLS; k += 32) {
        float v = lrow[k];
        if (v > m) { m = v; mi = k; }
    }
    for (int off = 16; off; off >>= 1) {
        float om = __shfl_xor(m, off);
        int   oi = __shfl_xor(mi, off);
        if (om > m || (om == m && oi < mi)) { m = om; mi = oi; }
    }

    // ---- base box (uniform across wave) ----
    const float4 bb = *(const float4*)(boxes_base + (size_t)row * 4);
    const float bx0 = bb.x - 0.5f * bb.z, by0 = bb.y - 0.5f * bb.w;
    const float bx1 = bb.x + 0.5f * bb.z, by1 = bb.y + 0.5f * bb.w;
    const float barea = (bx1 - bx0) * (by1 - by0);

    // ---- GIoU filter vs targets: lane t handles target t ----
    int ok = 1;
    if (lane < NT) {
        float4 tb = *(const float4*)(tboxes + ((size_t)b * NT + lane) * 4);
        float tx0 = tb.x - 0.5f * tb.z, ty0 = tb.y - 0.5f * tb.w;
        float tx1 = tb.x + 0.5f * tb.z, ty1 = tb.y + 0.5f * tb.w;
        float tarea = (tx1 - tx0) * (ty1 - ty0);
        float ltx = fmaxf(bx0, tx0), lty = fmaxf(by0, ty0);
        float rbx = fminf(bx1, tx1), rby = fminf(by1, ty1);
        float iw = fmaxf(rbx - ltx, 0.0f), ih = fmaxf(rby - lty, 0.0f);
        float inter = iw * ih;
        float uni = barea + tarea - inter;
        float iou = inter / uni;
        float cx0 = fminf(bx0, tx0), cy0 = fminf(by0, ty0);
        float cx1 = fmaxf(bx1, tx1), cy1 = fmaxf(by1, ty1);
        float cw = fmaxf(cx1 - cx0, 0.0f), ch = fmaxf(cy1 - cy0, 0.0f);
        float ac = cw * ch;
        float giou = iou - (ac - uni) / ac;
        ok = (-giou > -0.1f) ? 1 : 0;       // BBOX_THRESH = -0.1
    }
    const int keep_giou = __all(ok);

    if (lane == 0) {
        // keep_prob: sigmoid(max_logit) > 0.5  <=>  max_logit > 0  (exact)
        const int mask = (m > 0.0f) && keep_giou;
        ws[0 * NROW + row] = bb.x;
        ws[1 * NROW + row] = bb.y;
        ws[2 * NROW + row] = bb.z;
        ws[3 * NROW + row] = bb.w;
        ws[4 * NROW + row] = bx0;
        ws[5 * NROW + row] = by0;
        ws[6 * NROW + row] = bx1;
        ws[7 * NROW + row] = by1;
        ws[8 * NROW + row] = barea;
        ((int*)ws)[9 * NROW + row] = mi;
        ws[10 * NROW + row] = mask ? 1.0f : 0.0f;
        out_mask[row] = mask ? 1.0f : 0.0f;
    }
}

// ---------------- Kernel 1b: class-cost table precompute --------------------
// F[b,q,c] = W_CLS * focal(sigmoid(logit[b,q,c])) — 2.62M elements, moves all
// exp/log work out of the 25.9M-element dense loop (10x fewer transcendentals)
__global__ __launch_bounds__(256) void classcost_kernel(
    const float* __restrict__ pred_logits,  // [B, Q, NCLS]
    float* __restrict__ F)                  // [B, Q, NCLS]
{
    const size_t idx = (size_t)blockIdx.x * 256 + threadIdx.x;
    if (idx >= NLOGIT) return;
    F[idx] = class_cost_from_logit(pred_logits[idx]);
}

// ---------------- Kernel 2: dense pairwise cost -----------------------------
// Block: 256 threads, lane owns column j (register j-struct), iterates a
// 60-row q-chunk staged into LDS via async global->LDS b128 copies.
// srcRow is [B,Q,NCLS]: the precomputed class-cost table (USEF=true) or the
// raw logits (USEF=false fallback when workspace is too small).
template <bool USEF>
__global__ __launch_bounds__(256) void cost_kernel(
    const float* __restrict__ srcRow,       // [B, Q, NCLS]
    const float* __restrict__ pred_boxes,   // [B, Q, 4]
    const float* __restrict__ ws,
    float* __restrict__ outC)               // [B, Q, QB]
{
    __shared__ float4 sQ[QCHUNK];

    const int tid = threadIdx.x;
    const int b   = blockIdx.z;
    const int q0  = blockIdx.y * QCHUNK;
    const int j   = blockIdx.x * JTILE + tid;
    const bool valid = (j < QB);
    const int  jj  = valid ? j : 0;
    const int  row = b * QB + jj;

    // per-lane pseudo-target struct (coalesced SoA loads)
    const float jcx = ws[0 * NROW + row], jcy = ws[1 * NROW + row];
    const float jw  = ws[2 * NROW + row], jh  = ws[3 * NROW + row];
    const float jx0 = ws[4 * NROW + row], jy0 = ws[5 * NROW + row];
    const float jx1 = ws[6 * NROW + row], jy1 = ws[7 * NROW + row];
    const float jarea = ws[8 * NROW + row];
    const int   jl  = ((const int*)ws)[9 * NROW + row];
    const float jmask = ws[10 * NROW + row];

    // stage this block's q-chunk of pred boxes into LDS (async, 16B/lane)
    if (tid < QCHUNK) {
        const void* g = (const void*)(pred_boxes + ((size_t)(b * QQ + q0 + tid)) * 4);
        async_copy_b128_to_lds(g, (void*)&sQ[tid]);
    }
    wait_async_zero();
    __syncthreads();

    const float* rbase = srcRow + ((size_t)(b * QQ + q0)) * NCLS;

    for (int qi = 0; qi < QCHUNK; ++qi) {
        if (tid == 0 && qi + 1 < QCHUNK) {
            __builtin_prefetch(rbase + (size_t)(qi + 1) * NCLS, 0, 1);
        }
        const float4 qb = sQ[qi];                   // LDS broadcast read

        // class cost: one gather from the row (L0-resident across 8 waves)
        const float v = rbase[(size_t)qi * NCLS + jl];
        float clsW;
        if constexpr (USEF) {
            clsW = v;                               // already W_CLS * cost
        } else {
            clsW = class_cost_from_logit(v);
        }

        // L1 box cost (cxcywh)
        const float l1 = fabsf(qb.x - jcx) + fabsf(qb.y - jcy)
                       + fabsf(qb.z - jw)  + fabsf(qb.w - jh);

        // GIoU cost (xyxy)
        const float qx0 = qb.x - 0.5f * qb.z, qy0 = qb.y - 0.5f * qb.w;
        const float qx1 = qb.x + 0.5f * qb.z, qy1 = qb.y + 0.5f * qb.w;
        const float qarea = (qx1 - qx0) * (qy1 - qy0);
        const float ltx = fmaxf(qx0, jx0), lty = fmaxf(qy0, jy0);
        const float rbx = fminf(qx1, jx1), rby = fminf(qy1, jy1);
        const float iw = fmaxf(rbx - ltx, 0.0f), ih = fmaxf(rby - lty, 0.0f);
        const float inter = iw * ih;
        const float uni = qarea + jarea - inter;
        const float iou = inter / uni;
        const float cx0 = fminf(qx0, jx0), cy0 = fminf(qy0, jy0);
        const float cx1 = fmaxf(qx1, jx1), cy1 = fmaxf(qy1, jy1);
        const float cw = fmaxf(cx1 - cx0, 0.0f), ch = fmaxf(cy1 - cy0, 0.0f);
        const float ac = cw * ch;
        const float giou = iou - (ac - uni) / ac;

        const float cost = W_BOX * l1 + clsW + W_GIOU * (-giou);
        const float outv = (jmask != 0.0f) ? cost : BIGC;

        if (valid) {
            outC[((size_t)(b * QQ + q0 + qi)) * QB + j] = outv;
        }
    }
}

// ---------------- Launch ----------------------------------------------------

extern "C" void kernel_launch(void* const* d_in, const int* in_sizes, int n_in,
                              void* d_out, int out_size, void* d_ws, size_t ws_size,
                              hipStream_t stream) {
    const float* pred_logits = (const float*)d_in[0];  // [32,900,91]
    const float* pred_boxes  = (const float*)d_in[1];  // [32,900,4]
    const float* logits_base = (const float*)d_in[2];  // [32,900,91]
    const float* boxes_base  = (const float*)d_in[3];  // [32,900,4]
    const float* tboxes      = (const float*)d_in[4];  // [32,30,4]

    float* outC    = (float*)d_out;                    // [32,900,900]
    float* outMask = outC + (size_t)BB * QQ * QB;      // [32,900]
    float* ws      = (float*)d_ws;

    const size_t structBytes = (size_t)11 * NROW * sizeof(float);   // 1.27 MB
    const size_t fBytes      = NLOGIT * sizeof(float);              // 10.5 MB
    // deterministic path choice: ws_size is fixed for the run
    const bool useF = (ws_size >= structBytes + fBytes);

    // Phase 1: 28800 rows, one wave32 per row, 8 waves per 256-thread block
    prep_kernel<<<NROW / 8, 256, 0, stream>>>(logits_base, boxes_base, tboxes,
                                              ws, outMask);

    // Phase 2: 4 j-tiles x 15 q-chunks x 32 batches = 1920 blocks
    dim3 grid((QB + JTILE - 1) / JTILE, QQ / QCHUNK, BB);
    if (useF) {
        float* F = ws + (size_t)11 * NROW;
        classcost_kernel<<<(int)((NLOGIT + 255) / 256), 256, 0, stream>>>(pred_logits, F);
        cost_kernel<true><<<grid, 256, 0, stream>>>(F, pred_boxes, ws, outC);
    } else {
        cost_kernel<false><<<grid, 256, 0, stream>>>(pred_logits, pred_boxes, ws, outC);
    }
}